// NeighborAggregator_64398739637009
// MI455X (gfx1250) — compile-verified
//
#include <hip/hip_runtime.h>
#include <math.h>

typedef __attribute__((ext_vector_type(2))) float v2f;
typedef __attribute__((ext_vector_type(8))) float v8f;

#define SM_THREADS 1024
#define SM_MAXN    8192

__global__ void na_zero_kernel(float* __restrict__ p, int n) {
    int i = blockIdx.x * blockDim.x + threadIdx.x;
    if (i < n) p[i] = 0.0f;
}

__device__ __forceinline__ void atomic_add_f32(float* p, float v) {
    // HW global_atomic_add_f32 (coarse-grained device memory is fine here)
    unsafeAtomicAdd(p, v);
}

__global__ void na_edge_kernel(const float* __restrict__ data,
                               const int* __restrict__ rows,
                               const int* __restrict__ cols,
                               const float* __restrict__ vals,
                               float* __restrict__ A_raw,
                               int E, int N) {
    int e = blockIdx.x * blockDim.x + threadIdx.x;
    if (e >= E) return;
    int r = rows[e];
    int c = cols[e];
    float v = data[(size_t)r * (size_t)N + (size_t)c] * vals[e];
    atomic_add_f32(&A_raw[r], v);
}

// Single-block softmax over N (<= 8192) elements.
// Sum-of-exp reduction runs through V_WMMA_F32_16X16X4_F32 with an all-ones
// B matrix: D[m][n] = sum_k A[m][k] + C[m][n]  (exact f32 FMA math).
__global__ __launch_bounds__(SM_THREADS)
void na_softmax_kernel(const float* __restrict__ A_raw,
                       float* __restrict__ alpha, int N) {
    __shared__ float s_e[SM_MAXN];
    __shared__ float s_red[32];
    __shared__ float s_scalar;

    const int t    = threadIdx.x;
    const int lane = t & 31;
    const int wave = t >> 5;

    // ---- Phase 1: global max ----
    float m = -INFINITY;
    for (int i = t; i < N; i += SM_THREADS) m = fmaxf(m, A_raw[i]);
    #pragma unroll
    for (int off = 16; off >= 1; off >>= 1)
        m = fmaxf(m, __shfl_xor(m, off, 32));
    if (lane == 0) s_red[wave] = m;
    __syncthreads();
    if (t == 0) {
        float mm = s_red[0];
        for (int w = 1; w < 32; ++w) mm = fmaxf(mm, s_red[w]);
        s_scalar = mm;
    }
    __syncthreads();
    const float gmax = s_scalar;

    // ---- Phase 2: exp into LDS ----
    for (int i = t; i < N; i += SM_THREADS) s_e[i] = expf(A_raw[i] - gmax);
    __syncthreads();

    // ---- Phase 3: sum via chained f32 WMMA (wave covers 256 elements) ----
    const int base = wave * 256;
    v8f acc = {};
    v2f ones;
    ones[0] = 1.0f;
    ones[1] = 1.0f;
    #pragma unroll
    for (int it = 0; it < 4; ++it) {
        v2f a;
        a[0] = s_e[base + it * 64 + 2 * lane];
        a[1] = s_e[base + it * 64 + 2 * lane + 1];
        // (neg_a, A, neg_b, B, c_mod, C, reuse_a, reuse_b)
        acc = __builtin_amdgcn_wmma_f32_16x16x4_f32(
            false, a, false, ones, (short)0, acc, false, false);
    }
    float s = acc[0] + acc[1] + acc[2] + acc[3] +
              acc[4] + acc[5] + acc[6] + acc[7];   // sum over M=0..7 (or 8..15) of this column
    s += __shfl_xor(s, 16, 32);                    // combine the two M halves -> full wave sum
    if (lane == 0) s_red[wave] = s;
    __syncthreads();
    if (t == 0) {
        float total = 0.0f;
        for (int w = 0; w < 32; ++w) total += s_red[w];
        s_scalar = total;
    }
    __syncthreads();

    // ---- Phase 4: normalize ----
    const float inv = 1.0f / s_scalar;
    for (int i = t; i < N; i += SM_THREADS) alpha[i] = s_e[i] * inv;
}

extern "C" void kernel_launch(void* const* d_in, const int* in_sizes, int n_in,
                              void* d_out, int out_size, void* d_ws, size_t ws_size,
                              hipStream_t stream) {
    const float* data = (const float*)d_in[0];
    const int*   rows = (const int*)d_in[1];
    const int*   cols = (const int*)d_in[2];
    const float* vals = (const float*)d_in[3];

    const int E = in_sizes[1];
    const int N = out_size / 2;               // outputs: [alpha(N) ; A_raw(N)]

    float* alpha = (float*)d_out;
    float* A_raw = (float*)d_out + N;

    na_zero_kernel<<<(N + 255) / 256, 256, 0, stream>>>(A_raw, N);
    na_edge_kernel<<<(E + 255) / 256, 256, 0, stream>>>(data, rows, cols, vals,
                                                        A_raw, E, N);
    na_softmax_kernel<<<1, SM_THREADS, 0, stream>>>(A_raw, alpha, N);
}